// GNNModel_71811853189815
// MI455X (gfx1250) — compile-verified
//
#include <hip/hip_runtime.h>
#include <hip/hip_bf16.h>

typedef __attribute__((ext_vector_type(2))) float v2f;
typedef __attribute__((ext_vector_type(8))) float v8f;

#define LDSP 132  // padded LDS row stride (floats) to spread banks

__device__ __forceinline__ void atomAddF(float* p, float v) {
  // native global_atomic_add_f32 path
  unsafeAtomicAdd(p, v);
}

__device__ __forceinline__ v8f zero8() {
  v8f z = {0.f, 0.f, 0.f, 0.f, 0.f, 0.f, 0.f, 0.f};
  return z;
}

// ---------------- utility kernels ----------------

__global__ void zero_kernel(float* __restrict__ p, long n) {
  long i = (long)blockIdx.x * blockDim.x + threadIdx.x;
  if (i < n) p[i] = 0.f;
}

__global__ void degree_kernel(const int* __restrict__ src, const int* __restrict__ dst,
                              float* __restrict__ degs, float* __restrict__ degd, int E) {
  int e = blockIdx.x * blockDim.x + threadIdx.x;
  if (e < E) {
    atomAddF(&degs[src[e]], 1.f);
    atomAddF(&degd[dst[e]], 1.f);
  }
}

__global__ void norm_kernel(float* __restrict__ p, int n) {
  int i = blockIdx.x * blockDim.x + threadIdx.x;
  if (i < n) {
    float d = p[i];
    p[i] = (d > 0.5f) ? rsqrtf(d) : 0.f;
  }
}

// h[i, 0:64] = inv[i,:] * ns[i] ; h[i, 64:128] = pri[i,:] * ns[i]
__global__ void concat_scale_kernel(const float* __restrict__ inv, const float* __restrict__ pri,
                                    const float* __restrict__ ns, float* __restrict__ h, int N) {
  int idx = blockIdx.x * blockDim.x + threadIdx.x;
  if (idx < N * 32) {
    int node = idx >> 5, q = idx & 31;
    float s = ns[node];
    const float* sp = (q < 16) ? (inv + (size_t)node * 64 + q * 4)
                               : (pri + (size_t)node * 64 + (q - 16) * 4);
    float4 v = *(const float4*)sp;
    float* dp = h + (size_t)node * 128 + q * 4;
    dp[0] = v.x * s; dp[1] = v.y * s; dp[2] = v.z * s; dp[3] = v.w * s;
  }
}

// one wave per edge, 4 floats per lane (128 feats). L2-resident gather + fp atomics.
__global__ void spmm128_kernel(const float* __restrict__ h, const int* __restrict__ src,
                               const int* __restrict__ dst, float* __restrict__ agg, int E) {
  long idx = (long)blockIdx.x * blockDim.x + threadIdx.x;
  long e = idx >> 5;
  int lane = (int)(idx & 31);
  if (e < E) {
    int s = src[e], d = dst[e];
    float4 v = *(const float4*)(h + (size_t)s * 128 + lane * 4);
    float* dp = agg + (size_t)d * 128 + lane * 4;
    atomAddF(dp + 0, v.x);
    atomAddF(dp + 1, v.y);
    atomAddF(dp + 2, v.z);
    atomAddF(dp + 3, v.w);
  }
}

// 32-feature propagate: thread per (edge, feat)
__global__ void spmm32_kernel(const float* __restrict__ y, const int* __restrict__ src,
                              const int* __restrict__ dst, float* __restrict__ out, int E) {
  long idx = (long)blockIdx.x * blockDim.x + threadIdx.x;
  long e = idx >> 5;
  int f = (int)(idx & 31);
  if (e < E) {
    float v = y[(size_t)src[e] * 32 + f];
    atomAddF(out + (size_t)dst[e] * 32 + f, v);
  }
}

__global__ void finalize_kernel(float* __restrict__ out, const float* __restrict__ nd,
                                const float* __restrict__ b2, int N) {
  int idx = blockIdx.x * blockDim.x + threadIdx.x;
  if (idx < N * 32) {
    int i = idx >> 5, f = idx & 31;
    out[idx] = out[idx] * nd[i] + b2[f];
  }
}

// ---------------- fused double-GEMM (WMMA f32 16x16x4) ----------------
// Per 64-row tile:  x1 = relu((agg1*nd) @ W1 + b1);  y2 = (x1*ns) @ W2
// A fragment (lane l): M = l&15, K = kbase + 2*(l>>4) + {0,1}
// B fragment (lane l): K = kbase + 2*(l>>4) + {0,1}, N = nbase + (l&15)
// C/D (vgpr v, lane l): M = v + 8*(l>>4), N = nbase + (l&15)
__global__ __launch_bounds__(128) void gcn_gemm_fused(
    const float* __restrict__ agg1, const float* __restrict__ ns,
    const float* __restrict__ nd, const float* __restrict__ W1,
    const float* __restrict__ b1, const float* __restrict__ W2,
    float* __restrict__ y2, int N) {
  __shared__ float sX[64 * LDSP];
  __shared__ float sB1[128];

  const int tid = threadIdx.x;
  const int wave = tid >> 5;
  const int lane = tid & 31;
  const int m = lane & 15;
  const int half = lane >> 4;
  const int rowblk = blockIdx.x * 64;
  const int wrow = rowblk + wave * 16;

  if (tid < 128) sB1[tid] = b1[tid];

  // stage 64x128 tile of agg1 scaled by norm_dst (coalesced float4)
  for (int i = tid; i < 64 * 32; i += 128) {
    int r = i >> 5, c4 = i & 31;
    int gr = rowblk + r;
    float4 v = make_float4(0.f, 0.f, 0.f, 0.f);
    float s = 0.f;
    if (gr < N) {
      s = nd[gr];
      v = *(const float4*)(agg1 + (size_t)gr * 128 + c4 * 4);
    }
    float* dp = &sX[r * LDSP + c4 * 4];
    dp[0] = v.x * s; dp[1] = v.y * s; dp[2] = v.z * s; dp[3] = v.w * s;
  }
  __syncthreads();

  // --- GEMM1: [16x128] @ [128x128], K in steps of 4, 8 column tiles ---
  v8f c[8];
  for (int nt = 0; nt < 8; ++nt) c[nt] = zero8();
  for (int ks = 0; ks < 32; ++ks) {
    const int koff = ks * 4 + half * 2;
    const float* ap = &sX[(wave * 16 + m) * LDSP + koff];
    v2f a = {ap[0], ap[1]};
#pragma unroll
    for (int nt = 0; nt < 8; ++nt) {
      v2f b = {W1[(koff + 0) * 128 + nt * 16 + m],
               W1[(koff + 1) * 128 + nt * 16 + m]};
      c[nt] = __builtin_amdgcn_wmma_f32_16x16x4_f32(false, a, false, b,
                                                    (short)0, c[nt], false, false);
    }
  }
  __syncthreads();

  // epilogue: +b1, relu, * norm_src; re-layout C->A via LDS
#pragma unroll
  for (int v = 0; v < 8; ++v) {
    int r = v + half * 8;
    int gr = wrow + r;
    float s = (gr < N) ? ns[gr] : 0.f;
#pragma unroll
    for (int nt = 0; nt < 8; ++nt) {
      float x = c[nt][v] + sB1[nt * 16 + m];
      x = fmaxf(x, 0.f) * s;
      sX[(wave * 16 + r) * LDSP + nt * 16 + m] = x;
    }
  }
  __syncthreads();

  // --- GEMM2: [16x128] @ [128x32] ---
  v8f d0 = zero8(), d1 = zero8();
  for (int ks = 0; ks < 32; ++ks) {
    const int koff = ks * 4 + half * 2;
    const float* ap = &sX[(wave * 16 + m) * LDSP + koff];
    v2f a = {ap[0], ap[1]};
    v2f b0 = {W2[(koff + 0) * 32 + m], W2[(koff + 1) * 32 + m]};
    v2f b1v = {W2[(koff + 0) * 32 + 16 + m], W2[(koff + 1) * 32 + 16 + m]};
    d0 = __builtin_amdgcn_wmma_f32_16x16x4_f32(false, a, false, b0,
                                               (short)0, d0, false, false);
    d1 = __builtin_amdgcn_wmma_f32_16x16x4_f32(false, a, false, b1v,
                                               (short)0, d1, false, false);
  }

#pragma unroll
  for (int v = 0; v < 8; ++v) {
    int gr = wrow + v + half * 8;
    if (gr < N) {
      y2[(size_t)gr * 32 + m] = d0[v];
      y2[(size_t)gr * 32 + 16 + m] = d1[v];
    }
  }
}

// ---------------- launcher ----------------

extern "C" void kernel_launch(void* const* d_in, const int* in_sizes, int n_in,
                              void* d_out, int out_size, void* d_ws, size_t ws_size,
                              hipStream_t stream) {
  const float* inv = (const float*)d_in[0];
  const float* pri = (const float*)d_in[1];
  const float* W1 = (const float*)d_in[2];
  const float* b1 = (const float*)d_in[3];
  const float* W2 = (const float*)d_in[4];
  const float* b2 = (const float*)d_in[5];
  const int* src = (const int*)d_in[6];
  const int* dst = (const int*)d_in[7];
  float* out = (float*)d_out;

  const int N = in_sizes[0] / 64;  // 50000
  const int E = in_sizes[6];       // 800000

  float* ws = (float*)d_ws;
  float* ns = ws;                            // N   (degree_out -> norm_src)
  float* nd = ws + N;                        // N   (degree_in  -> norm_dst)
  float* h_in = ws + 2 * (size_t)N;          // N*128, reused as y2 later
  float* agg1 = h_in + (size_t)N * 128;      // N*128
  float* y2 = h_in;                          // alias (h_in dead after spmm1)

  const int B = 256;
  zero_kernel<<<(int)((2L * N + B - 1) / B), B, 0, stream>>>(ns, 2L * N);
  zero_kernel<<<(int)(((long)N * 128 + B - 1) / B), B, 0, stream>>>(agg1, (long)N * 128);
  zero_kernel<<<(int)(((long)N * 32 + B - 1) / B), B, 0, stream>>>(out, (long)N * 32);

  degree_kernel<<<(E + B - 1) / B, B, 0, stream>>>(src, dst, ns, nd, E);
  norm_kernel<<<(2 * N + B - 1) / B, B, 0, stream>>>(ns, 2 * N);

  concat_scale_kernel<<<(N * 32 + B - 1) / B, B, 0, stream>>>(inv, pri, ns, h_in, N);
  spmm128_kernel<<<(int)(((long)E * 32 + B - 1) / B), B, 0, stream>>>(h_in, src, dst, agg1, E);

  gcn_gemm_fused<<<(N + 63) / 64, 128, 0, stream>>>(agg1, ns, nd, W1, b1, W2, y2, N);

  spmm32_kernel<<<(int)(((long)E * 32 + B - 1) / B), B, 0, stream>>>(y2, src, dst, out, E);
  finalize_kernel<<<(N * 32 + B - 1) / B, B, 0, stream>>>(out, nd, b2, N);
}